// Net_53360673685530
// MI455X (gfx1250) — compile-verified
//
#include <hip/hip_runtime.h>

// ---------------------------------------------------------------------------
// Dehaze transmission-map optimization (1024x1024x3, patch=7, 100 iters).
// Stencil-iterative workload: no matrix contractions exist, so no WMMA use;
// CDNA5 paths exercised instead: async global->LDS staging (ASYNCcnt),
// s_wait_asynccnt, global_prefetch_b8, wave32 blocks.
// ---------------------------------------------------------------------------

namespace {
constexpr int   WIMG = 1024;
constexpr int   PSZ  = 7;
constexpr int   HP   = WIMG - PSZ;    // 1017
constexpr int   WP   = WIMG - PSZ;    // 1017
constexpr int   NPIX = HP * WP;       // 1034289  (divisible by 3)
constexpr float RATE = 0.001f;
constexpr float EPSV = 1e-7f;
constexpr int   BLK  = 256;
constexpr int   NITR = 100;
}

// ---- CDNA5 async global->LDS staging (guarded; falls back to plain LDS) ----
#if defined(__gfx1250__)
#  ifdef __has_builtin
#    if __has_builtin(__builtin_amdgcn_global_load_async_to_lds_b32)
#      define ATHENA_ASYNC_LDS 1
#    endif
#    if __has_builtin(__builtin_amdgcn_s_wait_asynccnt)
#      define ATHENA_WAIT_ASYNC 1
#    endif
#  endif
#endif

__device__ __forceinline__ void stage_f32(float* lds_dst, const float* gsrc) {
#if defined(ATHENA_ASYNC_LDS)
  // copies 4 bytes from the global address gsrc to the LDS address lds_dst  (tracked by ASYNCcnt, bypasses VGPRs)
  __builtin_amdgcn_global_load_async_to_lds_b32(
      (__attribute__((address_space(1))) int*)gsrc,
      (__attribute__((address_space(3))) int*)lds_dst,
      0, 0);
#else
  *lds_dst = *gsrc;
#endif
}

__device__ __forceinline__ void stage_barrier() {
#if defined(ATHENA_ASYNC_LDS)
#  if defined(ATHENA_WAIT_ASYNC)
  __builtin_amdgcn_s_wait_asynccnt(0);
#  else
  // s_barrier does NOT cover ASYNCcnt; force the wait explicitly.
  asm volatile("s_wait_asynccnt 0x0" ::: "memory");
#  endif
#endif
  __syncthreads();
}

// ---------------------------------------------------------------------------
// t0 = tlb = max_c (1 - img[y+3, x+3, c] / A[c])
// ---------------------------------------------------------------------------
__global__ __launch_bounds__(BLK)
void k_init_t(const float* __restrict__ img, const float* __restrict__ A,
              float* __restrict__ t) {
  int n = blockIdx.x * BLK + threadIdx.x;
  if (n >= NPIX) return;
  int y = n / WP, x = n - y * WP;
  const float* p = img + ((y + 3) * WIMG + (x + 3)) * 3;
  float v0 = 1.0f - p[0] / A[0];
  float v1 = 1.0f - p[1] / A[1];
  float v2 = 1.0f - p[2] / A[2];
  t[n] = fmaxf(v0, fmaxf(v1, v2));
}

// ---------------------------------------------------------------------------
// sig = compute_sig(dehaze(t)).  The (3,N) flat reinterpretation means sig[n]
// needs F at flat indices 3n-1..3n+3, i.e. t[n-1],t[n],t[n+1] and 5 img vals.
// t line staged through LDS (async on CDNA5).
// ---------------------------------------------------------------------------
__global__ __launch_bounds__(BLK)
void k_sig(const float* __restrict__ t, const float* __restrict__ img,
           const float* __restrict__ A, float* __restrict__ sig) {
  __shared__ float st[BLK + 2];           // t[base-1 .. base+BLK]
  const int base = blockIdx.x * BLK;
  const int tid  = threadIdx.x;
  {
    int gi = base - 1 + tid;
    gi = min(max(gi, 0), NPIX - 1);
    stage_f32(&st[tid], &t[gi]);
    if (tid < 2) {
      int g2 = base - 1 + BLK + tid;
      g2 = min(max(g2, 0), NPIX - 1);
      stage_f32(&st[BLK + tid], &t[g2]);
    }
  }
  stage_barrier();

  const int n = base + tid;
  if (n >= NPIX) return;

  const float a0 = A[0], a1 = A[1], a2 = A[2];
  const float it_m1 = 1.0f / st[tid];       // 1/t[n-1] (clamped idx)
  const float it_0  = 1.0f / st[tid + 1];   // 1/t[n]
  const float it_p1 = 1.0f / st[tid + 2];   // 1/t[n+1] (clamped idx)

  // img addresses: flat pixel m -> (y,x) in the (hp,wp) crop of the 1024-wide img
  int y0 = n / WP, x0 = n - y0 * WP;
  const float* pc = img + (y0 * WIMG + x0) * 3;
  int m1 = max(n - 1, 0);          int ym = m1 / WP, xm = m1 - ym * WP;
  int p1 = min(n + 1, NPIX - 1);   int yp = p1 / WP, xp = p1 - yp * WP;
  float im_m1_c2 = img[(ym * WIMG + xm) * 3 + 2];
  float im_p1_c0 = img[(yp * WIMG + xp) * 3 + 0];

  // F(k) for k = 3n-1 .. 3n+3  (Fv[i] = F(3n-1+i))
  float Fv[5];
  Fv[0] = (im_m1_c2 - a2) * it_m1 + a2;
  Fv[1] = (pc[0]    - a0) * it_0  + a0;
  Fv[2] = (pc[1]    - a1) * it_0  + a1;
  Fv[3] = (pc[2]    - a2) * it_0  + a2;
  Fv[4] = (im_p1_c0 - a0) * it_p1 + a0;

  float gsum = 0.0f;
#pragma unroll
  for (int i = 0; i < 3; ++i) {
    int k = 3 * n + i;                 // < 3*NPIX, fits int
    int r = k / NPIX;                  // row of the (3,N) view
    int j = k - r * NPIX;
    float gv;
    if (j == 0)             gv = Fv[i + 2] - Fv[i + 1];   // left edge
    else if (j == NPIX - 1) gv = Fv[i + 1] - Fv[i];       // right edge
    else                    gv = 0.5f * (Fv[i + 2] - Fv[i]);
    gsum += gv * gv;
  }
  float l2 = sqrtf(gsum);
  float z  = 48.0f * (l2 - 0.1f);
  sig[n] = 1.0f / (1.0f + __expf(z));    // sigmoid(-48*(l2-0.1))
}

// ---------------------------------------------------------------------------
// t_out = w0 - RATE * grad(loss)(w0, sig), w0 = (t_in<=0 ? EPS : t_in)
// Closed-form adjoint of the padded dx/dy loss:
//   g = (2/w0)*[ dx(y,x-1)sig(y,x-1) - dx(y,x+1)sig(y,x+1)
//              + dy(y+1,x)sig(y+1,x) - dy(y-1,x)sig(y-1,x) ]
// ---------------------------------------------------------------------------
__device__ __forceinline__ float w0_at(const float* t, int idx) {
  float v = t[idx];
  return v <= 0.0f ? EPSV : v;
}

__global__ __launch_bounds__(BLK)
void k_grad(const float* __restrict__ t_in, const float* __restrict__ sig,
            float* __restrict__ t_out) {
  int n = blockIdx.x * BLK + threadIdx.x;
  if (n >= NPIX) return;
  int y = n / WP, x = n - y * WP;

  // prefetch the row two below (next stencil arm) into cache
  __builtin_prefetch((const void*)(t_in + min(n + 2 * WP, NPIX - 1)), 0, 3);

  float w0c  = w0_at(t_in, n);
  float lg_c = __logf(w0c);
  float lg_xm2 = (x >= 2)     ? __logf(w0_at(t_in, n - 2))      : 0.0f;
  float lg_xp2 = (x + 2 < WP) ? __logf(w0_at(t_in, n + 2))      : 0.0f;
  float lg_ym2 = (y >= 2)     ? __logf(w0_at(t_in, n - 2 * WP)) : 0.0f;
  float lg_yp2 = (y + 2 < HP) ? __logf(w0_at(t_in, n + 2 * WP)) : 0.0f;

  float acc = 0.0f;
  if (x >= 1)     acc += (lg_c   - lg_xm2) * sig[n - 1];    // +dx(y,x-1)*sig
  if (x + 1 < WP) acc -= (lg_xp2 - lg_c)   * sig[n + 1];    // -dx(y,x+1)*sig
  if (y + 1 < HP) acc += (lg_c   - lg_yp2) * sig[n + WP];   // +dy(y+1,x)*sig
  if (y >= 1)     acc -= (lg_ym2 - lg_c)   * sig[n - WP];   // -dy(y-1,x)*sig

  float g = (2.0f / w0c) * acc;
  t_out[n] = w0c - RATE * g;
}

// ---------------------------------------------------------------------------
// out = dehaze(t_final) = (img_c - A)/t + A
// ---------------------------------------------------------------------------
__global__ __launch_bounds__(BLK)
void k_dehaze(const float* __restrict__ t, const float* __restrict__ img,
              const float* __restrict__ A, float* __restrict__ out) {
  int n = blockIdx.x * BLK + threadIdx.x;
  if (n >= NPIX) return;
  int y = n / WP, x = n - y * WP;
  const float* p = img + (y * WIMG + x) * 3;
  float tv = t[n];
  out[3 * n + 0] = (p[0] - A[0]) / tv + A[0];
  out[3 * n + 1] = (p[1] - A[1]) / tv + A[1];
  out[3 * n + 2] = (p[2] - A[2]) / tv + A[2];
}

// ---------------------------------------------------------------------------
extern "C" void kernel_launch(void* const* d_in, const int* in_sizes, int n_in,
                              void* d_out, int out_size, void* d_ws, size_t ws_size,
                              hipStream_t stream) {
  (void)in_sizes; (void)n_in; (void)out_size; (void)ws_size;
  const float* img = (const float*)d_in[0];   // (1024,1024,3) f32
  const float* A   = (const float*)d_in[1];   // (3,) f32
  // d_in[2] = patch_size (==7), baked into compile-time constants.
  float* out = (float*)d_out;                 // (1017,1017,3) f32

  float* t0 = (float*)d_ws;                   // ping
  float* t1 = t0 + NPIX;                      // pong
  float* sg = t1 + NPIX;                      // sigma weights

  const int nb = (NPIX + BLK - 1) / BLK;

  k_init_t<<<nb, BLK, 0, stream>>>(img, A, t0);
  k_sig   <<<nb, BLK, 0, stream>>>(t0, img, A, sg);

  for (int it = 0; it < NITR; ++it) {
    float* tin  = (it & 1) ? t1 : t0;
    float* tout = (it & 1) ? t0 : t1;
    k_grad<<<nb, BLK, 0, stream>>>(tin, sg, tout);
    k_sig <<<nb, BLK, 0, stream>>>(tout, img, A, sg);
  }
  // NITR=100 (even): last write was to t0.
  k_dehaze<<<nb, BLK, 0, stream>>>(t0, img, A, out);
}